// MoE_57466662420620
// MI455X (gfx1250) — compile-verified
//
#include <hip/hip_runtime.h>
#include <cstdint>
#include <cstddef>

#define N_TOK 4096
#define DIM   1024
#define FFN   4096
#define NE    8
#define TOPK  2

typedef float v2f __attribute__((ext_vector_type(2)));
typedef float v8f __attribute__((ext_vector_type(8)));

// ---------------- workspace layout (bytes) ----------------
static constexpr size_t OFF_GATES = 0;                         // N*E floats   = 131072 B
static constexpr size_t OFF_CW    = OFF_GATES + 131072;        // N*2 floats   =  32768 B
static constexpr size_t OFF_IDX   = OFF_CW    + 32768;         // N*2 ints     =  32768 B
static constexpr size_t OFF_CNT   = OFF_IDX   + 32768;         // 8 ints       (pad 64)
static constexpr size_t OFF_CE    = OFF_CNT   + 64;            // 8 ints       (pad 64)
static constexpr size_t OFF_LIST  = OFF_CE    + 64;            // 8*8192 ints  = 262144 B
static constexpr size_t OFF_Y     = OFF_LIST  + 262144;        // 8192*1024 floats = 32 MB

__device__ __forceinline__ float fast_tanh(float v) {
#if __has_builtin(__builtin_amdgcn_tanhf)
  return __builtin_amdgcn_tanhf(v);
#else
  return tanhf(v);
#endif
}

// ---------------- kernels ----------------

__global__ void zero_cnt_kernel(int* __restrict__ cnt) {
  if (threadIdx.x < NE) cnt[threadIdx.x] = 0;
}

// One block per token; wave w computes logit for expert w.
__global__ __launch_bounds__(256) void gate_kernel(
    const float* __restrict__ x, const float* __restrict__ wg,
    float* __restrict__ gates, float* __restrict__ cw,
    int* __restrict__ topidx, int* __restrict__ cnt) {
  int n = blockIdx.x;
  int tid = threadIdx.x, wv = tid >> 5, lane = tid & 31;
  const float* xr = x  + (size_t)n  * DIM;
  const float* wr = wg + (size_t)wv * DIM;
  float s = 0.f;
  for (int d = lane; d < DIM; d += 32) s += xr[d] * wr[d];
  for (int off = 16; off > 0; off >>= 1) s += __shfl_xor(s, off, 32);
  __shared__ float lg[NE];
  if (lane == 0) lg[wv] = s;
  __syncthreads();
  if (tid == 0) {
    float m = lg[0];
    for (int e = 1; e < NE; e++) m = fmaxf(m, lg[e]);
    float g[NE]; float sum = 0.f;
    for (int e = 0; e < NE; e++) { g[e] = __expf(lg[e] - m); sum += g[e]; }
    float inv = 1.f / sum;
    int i0 = 0;
    for (int e = 1; e < NE; e++) if (g[e] > g[i0]) i0 = e;
    int i1 = (i0 == 0) ? 1 : 0;
    for (int e = 0; e < NE; e++) if (e != i0 && g[e] > g[i1]) i1 = e;
    for (int e = 0; e < NE; e++) gates[(size_t)n * NE + e] = g[e] * inv;
    float g0 = g[i0] * inv, g1 = g[i1] * inv, den = g0 + g1;
    cw[n * 2 + 0] = g0 / den;
    cw[n * 2 + 1] = g1 / den;
    topidx[n * 2 + 0] = i0;
    topidx[n * 2 + 1] = i1;
    atomicAdd(&cnt[i0], 1);  // integer: deterministic
  }
}

// Deterministic fixed-tree reduction for l_aux -> d_out[0]
__global__ __launch_bounds__(256) void laux_kernel(
    const float* __restrict__ gates, const int* __restrict__ cnt,
    float* __restrict__ out0) {
  int tid = threadIdx.x, e = tid >> 5, lane = tid & 31;
  float s = 0.f;
  for (int n = lane; n < N_TOK; n += 32) s += gates[(size_t)n * NE + e];
  for (int off = 16; off > 0; off >>= 1) s += __shfl_xor(s, off, 32);
  __shared__ float me[NE];
  if (lane == 0) me[e] = s;
  __syncthreads();
  if (tid == 0) {
    float l = 0.f;
    for (int i = 0; i < NE; i++)
      l += (me[i] / (float)N_TOK) * ((float)cnt[i] / (float)N_TOK);
    out0[0] = l * (float)NE;
  }
}

// One block per expert: order-preserving compaction of assignment ids (n*2+k).
__global__ __launch_bounds__(256) void route_kernel(
    const int* __restrict__ topidx, int* __restrict__ lists,
    int* __restrict__ count_e) {
  int e = blockIdx.x, tid = threadIdx.x;
  __shared__ int sc[256];
  int base = 0;
  int* lst = lists + (size_t)e * (N_TOK * TOPK);
  for (int a0 = 0; a0 < N_TOK * TOPK; a0 += 256) {
    int aid = a0 + tid;
    int f = (topidx[aid] == e) ? 1 : 0;
    sc[tid] = f;
    __syncthreads();
    for (int off = 1; off < 256; off <<= 1) {
      int v = (tid >= off) ? sc[tid - off] : 0;
      __syncthreads();
      sc[tid] += v;
      __syncthreads();
    }
    if (f) lst[base + sc[tid] - 1] = aid;
    int tot = sc[255];
    __syncthreads();
    base += tot;
  }
  if (tid == 0) count_e[e] = base;
}

// ---------------- grouped-GEMM FFN with fp32 WMMA ----------------
// M-tile = 32 tokens, 512 threads = 16 waves arranged as 2 row-tiles x 8 col-groups.
#define MT  32
#define XP  1026   // padded LDS row stride for x tile (bank-conflict free)
#define HP  132    // padded LDS row stride for h tile

__global__ __launch_bounds__(512) void ffn_kernel(
    const float* __restrict__ x,
    const float* __restrict__ w1, const float* __restrict__ b1,
    const float* __restrict__ w2, const float* __restrict__ b2,
    const int* __restrict__ lists, const int* __restrict__ count_e,
    const float* __restrict__ cw, float* __restrict__ ybuf) {
  int e = blockIdx.y;
  int cnt = count_e[e];
  int r0 = blockIdx.x * MT;
  if (r0 >= cnt) return;

  __shared__ float xs[MT * XP];
  __shared__ float hs[MT * HP];
  __shared__ int   s_tok[MT];
  __shared__ int   s_aid[MT];
  __shared__ float s_c[MT];

  int tid = threadIdx.x, wv = tid >> 5, lane = tid & 31;
  int mr = wv >> 3;           // row-tile 0/1 (16 rows each)
  int wc = wv & 7;            // column-group: 128 output cols
  int half = lane >> 4, ln = lane & 15;

  if (tid < MT) {
    int g = r0 + tid;
    int aid = lists[(size_t)e * (N_TOK * TOPK) + (g < cnt ? g : r0)];
    s_aid[tid] = (g < cnt) ? aid : -1;
    s_tok[tid] = aid >> 1;
    s_c[tid]   = (g < cnt) ? cw[aid] : 0.f;
  }
  __syncthreads();

  // stage MT token rows of x into LDS
  for (int i = tid; i < MT * DIM; i += 512) {
    int r = i >> 10, c = i & 1023;
    xs[r * XP + c] = x[(size_t)s_tok[r] * DIM + c];
  }

  const float* w1e = w1 + (size_t)e * DIM * FFN;
  const float* w2e = w2 + (size_t)e * FFN * DIM;
  const float* b1e = b1 + (size_t)e * FFN;
  const float* b2e = b2 + (size_t)e * DIM;

  // y accumulators: wave owns rows [mr*16,+16), cols [wc*128,+128) as 8 tiles
  v8f yacc[8];
  for (int t = 0; t < 8; t++) {
    float bv = b2e[wc * 128 + t * 16 + ln];
    yacc[t] = (v8f){bv, bv, bv, bv, bv, bv, bv, bv};
  }
  __syncthreads();

  int arow = mr * 16 + ln;     // A-fragment row for this lane

  for (int fc = 0; fc < FFN; fc += 128) {
    // --- up-proj: wave computes h rows [mr*16,+16), cols [fc+wc*16,+16) ---
    int fcol = fc + wc * 16 + ln;
    float b1v = b1e[fcol];
    v8f hacc = (v8f){b1v, b1v, b1v, b1v, b1v, b1v, b1v, b1v};
    const float* w1c = w1e + fcol;  // + k*FFN
#pragma unroll 4
    for (int k = 0; k < DIM; k += 4) {
      int kk = k + 2 * half;
      v2f a; a.x = xs[arow * XP + kk]; a.y = xs[arow * XP + kk + 1];
      v2f b; b.x = w1c[(size_t)kk * FFN]; b.y = w1c[(size_t)(kk + 1) * FFN];
      hacc = __builtin_amdgcn_wmma_f32_16x16x4_f32(
          false, a, false, b, (short)0, hacc, false, false);
    }
    // gelu (tanh approx, matches jax.nn.gelu default) -> LDS
    __syncthreads();  // previous chunk's hs reads complete
#pragma unroll
    for (int r = 0; r < 8; r++) {
      float v = hacc[r];
      float g = 0.5f * v *
          (1.f + fast_tanh(0.7978845608028654f * (v + 0.044715f * v * v * v)));
      hs[(mr * 16 + r + 8 * half) * HP + wc * 16 + ln] = g;
    }
    __syncthreads();
    // --- down-proj: accumulate y += h_chunk @ w2[fc:fc+128, :] ---
#pragma unroll 2
    for (int k = 0; k < 128; k += 4) {
      int kk = k + 2 * half;
      v2f a; a.x = hs[arow * HP + kk]; a.y = hs[arow * HP + kk + 1];
      const float* w2r0 = w2e + (size_t)(fc + kk) * DIM;
      const float* w2r1 = w2r0 + DIM;
#pragma unroll
      for (int t = 0; t < 8; t++) {
        int dc = wc * 128 + t * 16 + ln;
        v2f b; b.x = w2r0[dc]; b.y = w2r1[dc];
        yacc[t] = __builtin_amdgcn_wmma_f32_16x16x4_f32(
            false, a, false, b, (short)0, yacc[t], false, false);
      }
    }
  }

  // store c * (h@w2 + b2) -> per-(token,slot) buffer; each slot written once
#pragma unroll
  for (int t = 0; t < 8; t++) {
#pragma unroll
    for (int r = 0; r < 8; r++) {
      int m = mr * 16 + r + 8 * half;
      if (s_aid[m] >= 0) {
        int col = wc * 128 + t * 16 + ln;
        ybuf[(size_t)s_aid[m] * DIM + col] = s_c[m] * yacc[t][r];
      }
    }
  }
}

__global__ __launch_bounds__(256) void combine_kernel(
    const float* __restrict__ ybuf, float* __restrict__ out) {
  size_t p = (size_t)blockIdx.x * 256 + threadIdx.x;  // p < N*D
  size_t n = p >> 10, d = p & 1023;
  out[p] = ybuf[(n * 2) * (size_t)DIM + d] + ybuf[(n * 2 + 1) * (size_t)DIM + d];
}

// ---------------- launch ----------------
extern "C" void kernel_launch(void* const* d_in, const int* in_sizes, int n_in,
                              void* d_out, int out_size, void* d_ws, size_t ws_size,
                              hipStream_t stream) {
  (void)in_sizes; (void)n_in; (void)out_size; (void)ws_size;
  const float* x  = (const float*)d_in[0];
  const float* wg = (const float*)d_in[1];
  const float* w1 = (const float*)d_in[2];
  const float* b1 = (const float*)d_in[3];
  const float* w2 = (const float*)d_in[4];
  const float* b2 = (const float*)d_in[5];
  float* out = (float*)d_out;  // out[0] = l_aux, out+1 = [N, D]

  char* ws = (char*)d_ws;
  float* gates   = (float*)(ws + OFF_GATES);
  float* cwp     = (float*)(ws + OFF_CW);
  int*   topidx  = (int*)  (ws + OFF_IDX);
  int*   cnt     = (int*)  (ws + OFF_CNT);
  int*   count_e = (int*)  (ws + OFF_CE);
  int*   lists   = (int*)  (ws + OFF_LIST);
  float* ybuf    = (float*)(ws + OFF_Y);

  zero_cnt_kernel<<<1, 64, 0, stream>>>(cnt);
  gate_kernel<<<N_TOK, 256, 0, stream>>>(x, wg, gates, cwp, topidx, cnt);
  laux_kernel<<<1, 256, 0, stream>>>(gates, cnt, out);
  route_kernel<<<NE, 256, 0, stream>>>(topidx, lists, count_e);
  ffn_kernel<<<dim3((N_TOK * TOPK) / MT, NE), 512, 0, stream>>>(
      x, w1, b1, w2, b2, lists, count_e, cwp, ybuf);
  combine_kernel<<<(N_TOK * DIM) / 256, 256, 0, stream>>>(ybuf, out + 1);
}